// RegularSparseLocalLayer_51642686767264
// MI455X (gfx1250) — compile-verified
//
#include <hip/hip_runtime.h>

typedef __attribute__((ext_vector_type(2))) float v2f;
typedef __attribute__((ext_vector_type(8))) float v8f;

#define NF     4096
#define DHALF  64
#define BAND   144        // 16 + 2*64, K extent per 16-wide out tile
#define BPAD   148        // padded LDS row length (592B: 16B aligned, conflict-free)
#define WAVES  4          // waves per block (wave32)
#define BROWS  (WAVES*16) // batch rows per block

__global__ __launch_bounds__(WAVES*32)
void banded_linear_wmma(const float* __restrict__ x,
                        const float* __restrict__ w,
                        const float* __restrict__ bias,
                        float* __restrict__ y)
{
    __shared__ float sB[16][BPAD];          // masked W^T band tile: sB[n][k]
    __shared__ float sX[WAVES][16][BPAD];   // per-wave x tile

    const int tid  = threadIdx.x;
    const int lane = tid & 31;
    const int wv   = tid >> 5;
    const int o0   = blockIdx.x << 4;       // out-feature tile base
    const int b0   = blockIdx.y * BROWS;    // batch tile base
    const int i0   = o0 - DHALF;            // first in-feature of the band (16-aligned)

    // ---- stage masked weight band: sB[n][k] = (|o-i|<=64 && i in range) ? W[o][i] : 0
    for (int t = tid; t < 16 * BAND; t += WAVES * 32) {
        int n = t / BAND;
        int k = t % BAND;                   // consecutive k -> coalesced W-row reads
        int i = i0 + k;
        float v = 0.f;
        if (i >= 0 && i < NF && k >= n && k <= n + 2 * DHALF)
            v = w[(size_t)(o0 + n) * NF + i];
        sB[n][k] = v;
    }

    // ---- stage this wave's 16 x-rows, cols i0..i0+143 (float4, zero OOB)
    {
        const int rowbase = b0 + wv * 16;
        for (int f = lane; f < 16 * (BAND / 4); f += 32) {
            int r  = f / (BAND / 4);
            int c4 = (f % (BAND / 4)) * 4;
            int i  = i0 + c4;
            const float* src = x + (size_t)(rowbase + r) * NF + i;
            float4 v;
            if (i >= 0 && i + 3 < NF) {
                v = *(const float4*)src;
            } else {
                v.x = (i + 0 >= 0 && i + 0 < NF) ? src[0] : 0.f;
                v.y = (i + 1 >= 0 && i + 1 < NF) ? src[1] : 0.f;
                v.z = (i + 2 >= 0 && i + 2 < NF) ? src[2] : 0.f;
                v.w = (i + 3 >= 0 && i + 3 < NF) ? src[3] : 0.f;
            }
            *(float4*)&sX[wv][r][c4] = v;
        }
    }

    __syncthreads();

    // ---- 16x16 tile via 36 x V_WMMA_F32_16X16X4_F32
    // fragment layout: element (vgpr v, lane L): K = v + 2*(L>=16), M/N = L%16
    const int m  = lane & 15;
    const int kh = (lane >> 4) << 1;        // 0 or 2
    const float* xr = &sX[wv][m][kh];
    const float* br = &sB[m][kh];

    v8f acc = {};
#pragma unroll
    for (int kk = 0; kk < BAND; kk += 4) {
        v2f a = *(const v2f*)(xr + kk);     // ds_load_b64, conflict-free
        v2f b = *(const v2f*)(br + kk);
        acc = __builtin_amdgcn_wmma_f32_16x16x4_f32(
                  false, a, false, b, (short)0, acc, false, false);
    }

    // ---- bias + store (C/D layout: vgpr j -> rows j and j+8; N = lane%16)
    const float bv = bias[o0 + m];
    const int rtop = b0 + wv * 16 + ((lane < 16) ? 0 : 8);
    float* yp = y + (size_t)rtop * NF + o0 + m;
#pragma unroll
    for (int j = 0; j < 8; ++j)
        yp[(size_t)j * NF] = acc[j] + bv;
}

extern "C" void kernel_launch(void* const* d_in, const int* in_sizes, int n_in,
                              void* d_out, int out_size, void* d_ws, size_t ws_size,
                              hipStream_t stream)
{
    const float* x    = (const float*)d_in[0];
    const float* w    = (const float*)d_in[1];
    const float* bias = (const float*)d_in[2];
    // d_in[3] (mask) is encoded structurally: |o - i| <= 64
    float* y = (float*)d_out;

    const int batch = in_sizes[0] / NF;     // 8192
    dim3 grid(NF / 16, batch / BROWS, 1);   // (256, 128)
    dim3 block(WAVES * 32, 1, 1);           // 128 threads = 4 waves
    hipLaunchKernelGGL(banded_linear_wmma, grid, block, 0, stream, x, w, bias, y);
}